// GCN_57956288692668
// MI455X (gfx1250) — compile-verified
//
#include <hip/hip_runtime.h>
#include <hip/hip_bf16.h>
#include <math.h>

typedef float v2f __attribute__((ext_vector_type(2)));
typedef float v8f __attribute__((ext_vector_type(8)));

#define GCN_N_NODES 50000

// ---------------- degree / normalization ----------------

__global__ void gcn_init_deg(float* __restrict__ deg, int n) {
    int i = blockIdx.x * blockDim.x + threadIdx.x;
    if (i < n) deg[i] = 1.0f;  // self-loop contribution
}

__global__ void gcn_deg_accum(const int* __restrict__ dst, float* __restrict__ deg, int E) {
    int e = blockIdx.x * blockDim.x + threadIdx.x;
    if (e < E) atomicAdd(&deg[dst[e]], 1.0f);
}

__global__ void gcn_rsqrt(float* __restrict__ deg, int n) {
    int i = blockIdx.x * blockDim.x + threadIdx.x;
    if (i < n) {
        float d = deg[i];
        deg[i] = (d > 0.0f) ? rsqrtf(d) : 0.0f;
    }
}

// ---------------- GEMM via V_WMMA_F32_16X16X4_F32 ----------------
// Y[M,N] = X[M,K] @ W[K,N], N = NT*16, K % 4 == 0, M % 16 == 0.
// One wave computes a 16-row tile across all N columns.
// A 16x4 f32 layout: lane L holds row M=L&15, K-pair (L>>4)*2 in {v0,v1}.
// B 4x16 f32 layout mirrored; C/D: VGPR j = rows {j, j+8}, N = lane&15.
template <int K, int NT>
__global__ __launch_bounds__(256) void gcn_gemm_wmma(const float* __restrict__ X,
                                                     const float* __restrict__ W,
                                                     float* __restrict__ Y, int M) {
    constexpr int N = NT * 16;
    const int wave = threadIdx.x >> 5;
    const int lane = threadIdx.x & 31;
    const int mtile = blockIdx.x * 8 + wave;   // wave-uniform
    if (mtile * 16 >= M) return;               // whole-wave exit: EXEC all-1s at WMMA

    const int row  = lane & 15;
    const int koff = (lane >> 4) << 1;         // 0 or 2
    const float* Arow = X + (size_t)(mtile * 16 + row) * K;

    v8f acc[NT] = {};

    for (int k = 0; k < K; k += 4) {
        v2f a;
        a.x = Arow[k + koff];
        a.y = Arow[k + koff + 1];
#pragma unroll
        for (int nt = 0; nt < NT; ++nt) {
            v2f b;
            b.x = W[(size_t)(k + koff) * N + nt * 16 + row];
            b.y = W[(size_t)(k + koff + 1) * N + nt * 16 + row];
            acc[nt] = __builtin_amdgcn_wmma_f32_16x16x4_f32(
                /*neg_a=*/false, a, /*neg_b=*/false, b,
                /*c_mod=*/(short)0, acc[nt], /*reuse_a=*/false, /*reuse_b=*/false);
        }
    }

    const int n = lane & 15;
    const int mhalf = (lane >> 4) << 3;        // 0 or 8
#pragma unroll
    for (int nt = 0; nt < NT; ++nt)
#pragma unroll
        for (int j = 0; j < 8; ++j)
            Y[(size_t)(mtile * 16 + mhalf + j) * N + nt * 16 + n] = acc[nt][j];
}

// ---------------- edge scatter (message aggregation) ----------------
// One wave per edge; F=64: each lane handles 2 features (float2 gather + 2 f32 atomics).
__global__ __launch_bounds__(256) void gcn_scatter64(const int* __restrict__ src,
                                                     const int* __restrict__ dst,
                                                     const float* __restrict__ dinv,
                                                     const float* __restrict__ H,
                                                     float* __restrict__ A, int E) {
    int t = blockIdx.x * blockDim.x + threadIdx.x;
    int e = t >> 5;
    int lane = t & 31;
    if (e >= E) return;
    int s = src[e], d = dst[e];
    float norm = dinv[s] * dinv[d];
    float2 hv = ((const float2*)(H + (size_t)s * 64))[lane];
    float* out = A + (size_t)d * 64 + lane * 2;
    atomicAdd(out,     hv.x * norm);
    atomicAdd(out + 1, hv.y * norm);
}

// F=32: one lane per feature.
__global__ __launch_bounds__(256) void gcn_scatter32(const int* __restrict__ src,
                                                     const int* __restrict__ dst,
                                                     const float* __restrict__ dinv,
                                                     const float* __restrict__ H,
                                                     float* __restrict__ A, int E) {
    int t = blockIdx.x * blockDim.x + threadIdx.x;
    int e = t >> 5;
    int f = t & 31;
    if (e >= E) return;
    int s = src[e], d = dst[e];
    float norm = dinv[s] * dinv[d];
    atomicAdd(A + (size_t)d * 32 + f, H[(size_t)s * 32 + f] * norm);
}

// ---------------- finalize (self-loop fold + activation) ----------------

__global__ void gcn_finalize_tanh64(const float* __restrict__ agg,
                                    const float* __restrict__ hlin,
                                    const float* __restrict__ dinv,
                                    float* __restrict__ h, int total) {
    int t = blockIdx.x * blockDim.x + threadIdx.x;
    if (t >= total) return;
    int i = t >> 6;  // /64
    float di = dinv[i];
    h[t] = tanhf(agg[t] + hlin[t] * di * di);
}

__global__ void gcn_finalize32(const float* __restrict__ agg,
                               const float* __restrict__ hlin,
                               const float* __restrict__ dinv,
                               float* __restrict__ o, int total) {
    int t = blockIdx.x * blockDim.x + threadIdx.x;
    if (t >= total) return;
    int i = t >> 5;  // /32
    float di = dinv[i];
    o[t] = agg[t] + hlin[t] * di * di;
}

// ---------------- launch ----------------

extern "C" void kernel_launch(void* const* d_in, const int* in_sizes, int n_in,
                              void* d_out, int out_size, void* d_ws, size_t ws_size,
                              hipStream_t stream) {
    const float* x  = (const float*)d_in[0];   // [50000,128]
    const int*   ei = (const int*)d_in[1];     // [2, E] flat
    const float* W1 = (const float*)d_in[2];   // [128,64]
    const float* W2 = (const float*)d_in[3];   // [64,32]

    const int N = GCN_N_NODES;
    const int E = in_sizes[1] / 2;
    const int* src = ei;
    const int* dst = ei + E;

    char* ws = (char*)d_ws;
    float* dinv  = (float*)(ws);                                        // 50000 f
    float* hlin  = (float*)(ws + 204800);                               // 50000*64 f
    float* agg1  = (float*)(ws + 204800 + 12800000);                    // 50000*64 f
    float* h2lin = (float*)(ws + 204800 + 2 * 12800000);                // 50000*32 f
    float* agg2  = (float*)(ws + 204800 + 2 * 12800000 + 6400000);      // 50000*32 f

    float* outp = (float*)d_out;                 // out: [50000,32]
    float* hout = outp + (size_t)N * 32;         // h:   [50000,64]

    // degrees -> dinv (in place)
    gcn_init_deg<<<(N + 255) / 256, 256, 0, stream>>>(dinv, N);
    gcn_deg_accum<<<(E + 255) / 256, 256, 0, stream>>>(dst, dinv, E);
    gcn_rsqrt<<<(N + 255) / 256, 256, 0, stream>>>(dinv, N);

    hipMemsetAsync(agg1, 0, (size_t)N * 64 * sizeof(float), stream);
    hipMemsetAsync(agg2, 0, (size_t)N * 32 * sizeof(float), stream);

    const int mtiles = N / 16;                   // 3125 (N % 16 == 0)
    const int gemm_blocks = (mtiles + 7) / 8;

    // Layer 1
    gcn_gemm_wmma<128, 4><<<gemm_blocks, 256, 0, stream>>>(x, W1, hlin, N);
    {
        long long threads = (long long)E * 32;
        int blocks = (int)((threads + 255) / 256);
        gcn_scatter64<<<blocks, 256, 0, stream>>>(src, dst, dinv, hlin, agg1, E);
    }
    gcn_finalize_tanh64<<<(N * 64 + 255) / 256, 256, 0, stream>>>(agg1, hlin, dinv, hout, N * 64);

    // Layer 2
    gcn_gemm_wmma<64, 2><<<gemm_blocks, 256, 0, stream>>>(hout, W2, h2lin, N);
    {
        long long threads = (long long)E * 32;
        int blocks = (int)((threads + 255) / 256);
        gcn_scatter32<<<blocks, 256, 0, stream>>>(src, dst, dinv, h2lin, agg2, E);
    }
    gcn_finalize32<<<(N * 32 + 255) / 256, 256, 0, stream>>>(agg2, h2lin, dinv, outp, N * 32);
}